// ContrastiveLoss_40596030882175
// MI455X (gfx1250) — compile-verified
//
#include <hip/hip_runtime.h>

// CDNA5 / gfx1250, wave32. Contrastive loss collapsed to O(N*D) reductions,
// with the per-row dot products computed as diagonals of WMMA f32 Gram tiles.

typedef __attribute__((ext_vector_type(2))) float v2f;
typedef __attribute__((ext_vector_type(8))) float v8f;

#define NROWS 4096
#define DIM   512

// Workspace layout (float offsets):
//   [0,512)             T (column sums of txt)
//   [512]               S_txt = sum_j ||txt_j||^2
//   [1024, 1024+32*512) T partials (32 row-chunks x 512 cols)
//   [17408, 17472)      S partials (64 blocks)
//   [17472, 17728)      per-block loss sums (256 blocks)
#define WS_T      0
#define WS_STXT   512
#define WS_TPART  1024
#define WS_SPART  17408
#define WS_BSUM   17472

// ---------------------------------------------------------------------------
// Pass 1: per-(col-chunk, row-chunk) partial column sums + partial sum of squares
// grid (2, 32), block 256. Coalesced: consecutive threads read consecutive cols.
// ---------------------------------------------------------------------------
__global__ void colsum_partials(const float* __restrict__ txt, float* __restrict__ ws) {
    __shared__ float red[256];
    const int col  = blockIdx.x * 256 + threadIdx.x;   // 0..511
    const int row0 = blockIdx.y * 128;
    const float* p = txt + (size_t)row0 * DIM + col;
    float tsum = 0.f, sq = 0.f;
#pragma unroll 4
    for (int r = 0; r < 128; ++r) {
        float v = p[(size_t)r * DIM];
        tsum += v;
        sq   += v * v;
    }
    ws[WS_TPART + blockIdx.y * DIM + col] = tsum;
    red[threadIdx.x] = sq;
    __syncthreads();
    for (int s = 128; s > 0; s >>= 1) {        // deterministic tree reduce
        if (threadIdx.x < s) red[threadIdx.x] += red[threadIdx.x + s];
        __syncthreads();
    }
    if (threadIdx.x == 0) ws[WS_SPART + blockIdx.y * 2 + blockIdx.x] = red[0];
}

// ---------------------------------------------------------------------------
// Pass 2: fold partials -> T[512] and S_txt. 1 block x 512 threads.
// ---------------------------------------------------------------------------
__global__ void colsum_final(float* __restrict__ ws) {
    const int c = threadIdx.x;                 // 0..511
    float s = 0.f;
#pragma unroll
    for (int b = 0; b < 32; ++b) s += ws[WS_TPART + b * DIM + c];
    ws[WS_T + c] = s;
    if (c == 0) {
        float q = 0.f;
        for (int i = 0; i < 64; ++i) q += ws[WS_SPART + i];
        ws[WS_STXT] = q;
    }
}

// ---------------------------------------------------------------------------
// Pass 3: main pass. One wave (32 threads) per 16-row tile; 256 blocks.
// Per K-step of 4 columns, each lane loads one float2 of img, one of txt,
// one of T (from LDS), then issues 4 x V_WMMA_F32_16X16X4_F32:
//   cII = img.img^T  (diag -> ||img_i||^2)
//   cTT = txt.txt^T  (diag -> ||txt_i||^2)
//   cIT = img.txt^T  (diag -> img_i . txt_i)
//   cIV = img.T^T    (diag -> img_i . T, B fragment broadcast from T)
// A-fragment layout (lane L = M=L%16, K pair 2*(L/16)) equals the B-fragment
// layout of the transposed operand, so A and B reuse the same loads.
// pos_i is computed elementwise as sum (a-b)^2 to match the reference exactly.
// ---------------------------------------------------------------------------
__global__ void tile_kernel(const float* __restrict__ img,
                            const float* __restrict__ txt,
                            float* __restrict__ ws) {
    __shared__ float Tlds[DIM];
    __shared__ float imgsqL[16], txtsqL[16], dotitL[16], dotivL[16];
    __shared__ float posL[32], contribL[16];

    const int L = threadIdx.x;                 // lane 0..31, full wave active
    for (int i = L; i < DIM; i += 32) Tlds[i] = ws[WS_T + i];
    __syncthreads();

    const int rowBase = blockIdx.x * 16;
    const int M       = L & 15;
    const int colOff  = (L >> 4) * 2;          // half 0 -> cols k,k+1 ; half 1 -> k+2,k+3
    const float* ip = img + (size_t)(rowBase + M) * DIM + colOff;
    const float* tp = txt + (size_t)(rowBase + M) * DIM + colOff;

    v8f cII = {}; v8f cTT = {}; v8f cIT = {}; v8f cIV = {};
    float posAcc = 0.f;

    for (int k = 0; k < DIM; k += 4) {
        v2f a = *(const v2f*)(ip + k);
        v2f b = *(const v2f*)(tp + k);
        v2f t; t.x = Tlds[k + colOff]; t.y = Tlds[k + colOff + 1];

        cII = __builtin_amdgcn_wmma_f32_16x16x4_f32(false, a, false, a, (short)0, cII, false, false);
        cTT = __builtin_amdgcn_wmma_f32_16x16x4_f32(false, b, false, b, (short)0, cTT, false, false);
        cIT = __builtin_amdgcn_wmma_f32_16x16x4_f32(false, a, false, b, (short)0, cIT, false, false);
        cIV = __builtin_amdgcn_wmma_f32_16x16x4_f32(false, a, false, t, (short)0, cIV, false, false);

        float d0 = a.x - b.x, d1 = a.y - b.y;
        posAcc += d0 * d0 + d1 * d1;
    }

    posL[L] = posAcc;
    // Diagonal C[d][d]: d<8 -> lane d, component d ; d>=8 -> lane d+8+... = d+16? no:
    // lanes 24..31 hold d = L-16, component L&7.
    if (L < 8 || L >= 24) {
        int row = (L < 8) ? L : (L - 16);
        int idx = L & 7;
        imgsqL[row] = cII[idx];
        txtsqL[row] = cTT[idx];
        dotitL[row] = cIT[idx];
        dotivL[row] = cIV[idx];
    }
    __syncthreads();

    if (L < 16) {
        float imgsq = imgsqL[L];
        float txtsq = txtsqL[L];
        float dotit = dotitL[L];
        float dotiv = dotivL[L];
        float pos   = posL[L] + posL[L + 16];
        float S_txt = ws[WS_STXT];
        // sum_j sq[i,j] = N*||img||^2 + S_txt - 2*img.T
        float rowsum = (float)NROWS * imgsq + S_txt - 2.0f * dotiv;
        float sqii   = imgsq + txtsq - 2.0f * dotit;
        float neg    = (rowsum - sqii) / (float)(NROWS - 1);
        float c      = pos - neg;
        contribL[L]  = c > 0.f ? c : 0.f;
    }
    __syncthreads();

    if (L == 0) {
        float s = 0.f;
#pragma unroll
        for (int i = 0; i < 16; ++i) s += contribL[i];
        ws[WS_BSUM + blockIdx.x] = s;
    }
}

// ---------------------------------------------------------------------------
// Pass 4: deterministic final reduction of the 256 block sums -> mean loss.
// ---------------------------------------------------------------------------
__global__ void final_reduce(const float* __restrict__ ws, float* __restrict__ out) {
    __shared__ float red[256];
    red[threadIdx.x] = ws[WS_BSUM + threadIdx.x];
    __syncthreads();
    for (int s = 128; s > 0; s >>= 1) {
        if (threadIdx.x < s) red[threadIdx.x] += red[threadIdx.x + s];
        __syncthreads();
    }
    if (threadIdx.x == 0) out[0] = red[0] / (float)NROWS;
}

extern "C" void kernel_launch(void* const* d_in, const int* in_sizes, int n_in,
                              void* d_out, int out_size, void* d_ws, size_t ws_size,
                              hipStream_t stream) {
    (void)in_sizes; (void)n_in; (void)out_size; (void)ws_size;
    const float* img = (const float*)d_in[0];
    const float* txt = (const float*)d_in[1];
    float* out = (float*)d_out;
    float* ws  = (float*)d_ws;

    colsum_partials<<<dim3(2, 32), 256, 0, stream>>>(txt, ws);
    colsum_final  <<<1, 512, 0, stream>>>(ws);
    tile_kernel   <<<256, 32, 0, stream>>>(img, txt, ws);
    final_reduce  <<<1, 256, 0, stream>>>(ws, out);
}